// Tokenizer_16097537426059
// MI455X (gfx1250) — compile-verified
//
#include <hip/hip_runtime.h>
#include <math.h>

typedef __attribute__((ext_vector_type(2))) float v2f;
typedef __attribute__((ext_vector_type(4))) float v4f;
typedef __attribute__((ext_vector_type(8))) float v8f;

#define TOK_PER_WG 32
#define C_DIM 256
#define V_DIM 2048
#define T_DIM 2048
#define B_DIM 16
#define ZA_STRIDE 260   // 256 + 4 pad: kills LDS bank conflicts on 16-row strided A loads
#define LG_STRIDE 2052  // 2048 + 4 pad: kills conflicts on D stores (rows r / r+8)

// Output layout (flat f32, concat in reference return order)
#define O_LP 0ull                    // log_probs: 16*2048*2048
#define O_ZQ 67108864ull             // z_q_st:    16*2048*256
#define O_SM 75497472ull             // smoothness_loss
#define O_CM 75497473ull             // commitment_loss
#define O_RL 75497474ull             // reinforce_loss
#define O_EM 75497475ull             // e_mean: 2048

// workspace: ws[0]=mask_sum, ws[1]=commit_num, ws[2]=smooth_num, ws[3..2050]=hist
#define WS_FLOATS 2051

__global__ void vq_zero(float* __restrict__ ws) {
    int i = blockIdx.x * 256 + threadIdx.x;
    if (i < WS_FLOATS) ws[i] = 0.0f;
}

__global__ __launch_bounds__(512)
void vq_main(const float* __restrict__ z, const float* __restrict__ mask,
             const float* __restrict__ e,
             float* __restrict__ log_probs, float* __restrict__ zq_st,
             float* __restrict__ ws)
{
    __shared__ float zA[TOK_PER_WG * ZA_STRIDE];   //  33,280 B
    __shared__ float lg[TOK_PER_WG * LG_STRIDE];   // 262,656 B
    __shared__ float lseArr[TOK_PER_WG];
    __shared__ int   topArr[TOK_PER_WG];

    const int tid  = threadIdx.x;
    const int w    = tid >> 5;      // wave id, 16 waves
    const int lane = tid & 31;
    const int half = lane >> 4;
    const int l15  = lane & 15;
    const int tokBase = blockIdx.x * TOK_PER_WG;

    // ---- Phase 0: cooperative load of z tile (32 x 256) into LDS ----
    // z is streamed once: non-temporal loads keep e resident in L2/WGP$.
    {
        const v4f* zg = (const v4f*)(z + (size_t)tokBase * C_DIM);
        for (int i = tid; i < TOK_PER_WG * (C_DIM / 4); i += 512) {
            int m  = i >> 6;     // 64 float4 per token row
            int k4 = i & 63;
            v4f v = __builtin_nontemporal_load(&zg[m * 64 + k4]);
            *(v4f*)&zA[m * ZA_STRIDE + k4 * 4] = v;
        }
    }
    __syncthreads();

    // ---- Phase 1: f32 WMMA GEMM: lg[m][v] = sum_k zA[m][k] * e[v][k] ----
    // Wave w: mtile = w>>3 (rows 16*mtile..+15); ntiles (w&7)+8t, t=0..15.
    {
        const int mtile = w >> 3;
        const float* zArow = &zA[(mtile * 16 + l15) * ZA_STRIDE];
        for (int t = 0; t < 16; ++t) {
            const int nt = (w & 7) + 8 * t;
            const float* erow = e + (size_t)(nt * 16 + l15) * C_DIM;
            v8f c = {};
            #pragma unroll
            for (int kk = 0; kk < 64; ++kk) {
                const int col = kk * 4 + 2 * half;           // 8B-aligned pair
                v2f a = *(const v2f*)(zArow + col);          // A: 16x4 tile
                v2f b = *(const v2f*)(erow + col);           // B: 4x16 tile (e row-major)
                c = __builtin_amdgcn_wmma_f32_16x16x4_f32(
                        false, a, false, b, (short)0, c, false, false);
            }
            // D layout: vgpr r -> row (mtile*16 + r + 8*half), col nt*16 + l15
            float* dst = &lg[(mtile * 16 + 8 * half) * LG_STRIDE + nt * 16 + l15];
            #pragma unroll
            for (int r = 0; r < 8; ++r) dst[r * LG_STRIDE] = c[r];
        }
    }
    __syncthreads();

    // ---- Phase 2: per-token argmax + logsumexp (wave w -> tokens 2w, 2w+1) ----
    for (int s = 0; s < 2; ++s) {
        const int m = 2 * w + s;
        const float* row = &lg[m * LG_STRIDE];
        float mx = -INFINITY; int mi = 0x7fffffff;
        for (int i = lane; i < V_DIM; i += 32) {
            float v = row[i];
            if (v > mx) { mx = v; mi = i; }   // within-lane: strict > keeps lowest idx
        }
        #pragma unroll
        for (int off = 16; off >= 1; off >>= 1) {
            float ov = __shfl_xor(mx, off);
            int   oi = __shfl_xor(mi, off);
            if (ov > mx || (ov == mx && oi < mi)) { mx = ov; mi = oi; }
        }
        float sum = 0.0f;
        for (int i = lane; i < V_DIM; i += 32) sum += expf(row[i] - mx);
        #pragma unroll
        for (int off = 16; off >= 1; off >>= 1) sum += __shfl_xor(sum, off);
        if (lane == 0) { lseArr[m] = mx + logf(sum); topArr[m] = mi; }
    }
    __syncthreads();

    // ---- Phase 3: write log_probs = logits - lse (coalesced, non-temporal) ----
    {
        for (int it = 0; it < 32; ++it) {
            int lin   = it * 512 + tid;          // 16384 float4 total
            int m     = lin >> 9;                // 512 float4 per token row
            int c4    = lin & 511;
            v4f v     = *(const v4f*)&lg[m * LG_STRIDE + c4 * 4];
            float lse = lseArr[m];
            v -= lse;
            __builtin_nontemporal_store(
                v, (v4f*)&log_probs[(size_t)(tokBase + m) * V_DIM + c4 * 4]);
        }
    }

    // ---- Phase 4: z_q_st, commitment, smoothness, histogram, mask_sum ----
    {
        float commAcc = 0.0f, smoothAcc = 0.0f, mkAcc = 0.0f;
        for (int s = 0; s < 2; ++s) {
            const int m = 2 * w + s;
            const int tokflat = tokBase + m;
            const float mk = mask[tokflat];
            const int top = topArr[m];
            const float* er = e + (size_t)top * C_DIM;
            const float* zr = &zA[m * ZA_STRIDE];
            const bool hasPrev = (tokflat % T_DIM) != 0;
            float comm = 0.0f, smooth = 0.0f;
            for (int q = 0; q < 8; ++q) {
                int k = lane + q * 32;
                float zv = zr[k];
                float ev = er[k];
                float d = zv - ev; comm += d * d;
                __builtin_nontemporal_store(
                    ev * mk, &zq_st[(size_t)tokflat * C_DIM + k]);
                if (hasPrev) {
                    float pz = (m > 0) ? zA[(m - 1) * ZA_STRIDE + k]
                                       : z[(size_t)(tokflat - 1) * C_DIM + k];
                    float sd = zv - pz; smooth += sd * sd;
                }
            }
            commAcc   += comm * mk;
            smoothAcc += smooth * mk;
            mkAcc     += mk;
            if (lane == 0) atomicAdd(&ws[3 + top], mk);   // histogram
        }
        #pragma unroll
        for (int off = 16; off >= 1; off >>= 1) {
            commAcc   += __shfl_xor(commAcc, off);
            smoothAcc += __shfl_xor(smoothAcc, off);
        }
        if (lane == 0) {
            atomicAdd(&ws[0], mkAcc);
            atomicAdd(&ws[1], commAcc);
            atomicAdd(&ws[2], smoothAcc);
        }
    }
}

__global__ void vq_finalize(const float* __restrict__ adv,
                            const float* __restrict__ ws,
                            float* __restrict__ out)
{
    int tid = threadIdx.x;
    float msum = ws[0];
    for (int v = tid; v < V_DIM; v += 256) out[O_EM + v] = ws[3 + v] / msum;
    if (tid == 0) {
        float valid = msum * (float)C_DIM;
        out[O_SM] = ws[2] / valid;
        out[O_CM] = ws[1] / valid;
        // coef == exp(0) == 1 => per_b = -(adv[b,0]+adv[b,1])/2 ; mean over b
        float r = 0.0f;
        for (int b = 0; b < B_DIM; ++b) r += -(adv[2 * b] + adv[2 * b + 1]) * 0.5f;
        out[O_RL] = r / (float)B_DIM;
    }
}

extern "C" void kernel_launch(void* const* d_in, const int* in_sizes, int n_in,
                              void* d_out, int out_size, void* d_ws, size_t ws_size,
                              hipStream_t stream) {
    (void)in_sizes; (void)n_in; (void)out_size; (void)ws_size;
    const float* z    = (const float*)d_in[0];   // (16,2048,256)
    const float* mask = (const float*)d_in[1];   // (16,2048,1)
    const float* e    = (const float*)d_in[2];   // (2048,256)
    const float* adv  = (const float*)d_in[3];   // (16,2)
    float* out = (float*)d_out;
    float* ws  = (float*)d_ws;

    vq_zero<<<(WS_FLOATS + 255) / 256, 256, 0, stream>>>(ws);
    vq_main<<<(B_DIM * T_DIM) / TOK_PER_WG, 512, 0, stream>>>(
        z, mask, e, out + O_LP, out + O_ZQ, ws);
    vq_finalize<<<1, 256, 0, stream>>>(adv, ws, out);
}